// GIN_41961830482646
// MI455X (gfx1250) — compile-verified
//
#include <hip/hip_runtime.h>

#define N_NODES_C 102400
#define NUM_ROI_C 200
#define HID_C     128
#define BATCH_C   512
#define N_EDGES_C 3276800
#define BN_EPS_C  1e-5f

typedef __attribute__((ext_vector_type(2))) float v2f;
typedef __attribute__((ext_vector_type(8))) float v8f;

// ---------------------------------------------------------------------------
// gfx1250 async global->LDS copy (16B per lane), tracked by ASYNCcnt.
// vdst = per-lane LDS byte address (low 32 bits of the generic pointer),
// vaddr = per-lane 64-bit global address, GV addressing mode.
// ---------------------------------------------------------------------------
__device__ __forceinline__ void async_copy_f4(const float* gsrc, float* ldst) {
  const unsigned la = (unsigned)(size_t)(void*)ldst;       // LDS offset
  const unsigned long long ga = (unsigned long long)(const void*)gsrc;
  asm volatile("global_load_async_to_lds_b128 %0, %1, off"
               :: "v"(la), "v"(ga) : "memory");
}

template <int N>
__device__ __forceinline__ void wait_asynccnt() {
  asm volatile("s_wait_asynccnt %0" :: "n"(N) : "memory");
}

// ---------------------------------------------------------------------------
// Zero-fill
// ---------------------------------------------------------------------------
__global__ void fill_zero_kernel(float* __restrict__ p, long long n) {
  long long i = (long long)blockIdx.x * blockDim.x + threadIdx.x;
  if (i < n) p[i] = 0.0f;
}

// ---------------------------------------------------------------------------
// Weight transpose + pad:  WT[Npad][Kpad] = W[K][N]^T, zero-filled pads.
// LDS-tiled 32x32 so both global accesses are coalesced.
// ---------------------------------------------------------------------------
__global__ void transpose_pad_kernel(const float* __restrict__ in,
                                     float* __restrict__ out,
                                     int K, int N, int Kpad, int Npad) {
  __shared__ float t[32][33];
  const int k0 = blockIdx.x * 32;
  const int n0 = blockIdx.y * 32;
  #pragma unroll
  for (int j = threadIdx.y; j < 32; j += 8) {
    const int k = k0 + j, n = n0 + threadIdx.x;
    t[j][threadIdx.x] = (k < K && n < N) ? in[(size_t)k * N + n] : 0.0f;
  }
  __syncthreads();
  #pragma unroll
  for (int j = threadIdx.y; j < 32; j += 8) {
    const int n = n0 + j, k = k0 + threadIdx.x;
    if (n < Npad && k < Kpad) out[(size_t)n * Kpad + k] = t[threadIdx.x][j];
  }
}

// ---------------------------------------------------------------------------
// Edge scatter-add: agg[dst[e]] += X[src[e]]  (feature dim F, F % 4 == 0)
// One thread per (edge, 4-feature group); hardware f32 atomics resolve at L2.
// ---------------------------------------------------------------------------
__global__ void scatter_add_kernel(const float* __restrict__ X,
                                   const int* __restrict__ src,
                                   const int* __restrict__ dst,
                                   float* __restrict__ agg,
                                   int F, long long nwork) {
  long long i = (long long)blockIdx.x * blockDim.x + threadIdx.x;
  if (i >= nwork) return;
  const int f4 = F >> 2;
  long long e = i / f4;
  int f = (int)(i - e * (long long)f4) << 2;
  const int s = src[e];
  const int d = dst[e];
  const float4 v = *reinterpret_cast<const float4*>(X + (size_t)s * F + f);
  float* p = agg + (size_t)d * F + f;
  __hip_atomic_fetch_add(p + 0, v.x, __ATOMIC_RELAXED, __HIP_MEMORY_SCOPE_AGENT);
  __hip_atomic_fetch_add(p + 1, v.y, __ATOMIC_RELAXED, __HIP_MEMORY_SCOPE_AGENT);
  __hip_atomic_fetch_add(p + 2, v.z, __ATOMIC_RELAXED, __HIP_MEMORY_SCOPE_AGENT);
  __hip_atomic_fetch_add(p + 3, v.w, __ATOMIC_RELAXED, __HIP_MEMORY_SCOPE_AGENT);
}

// ---------------------------------------------------------------------------
// fp32 WMMA GEMM:  Out[M,N] = op( (A1 [+ A2]) @ W + bias ), W given as
// pre-transposed padded WT[Npad][Kpad].
//   block = 128 threads = 4 wave32; block tile = 64 rows x 64 cols.
//   Each wave owns 16 rows and 4 accumulators across N: every A fragment
//   feeds 4x V_WMMA_F32_16X16X4_F32. Both LDS tiles are [64][20] K-major, so
//   every fragment pair is an adjacent ds b64 read (no repack movs).
//   Double-buffered stages filled by GLOBAL_LOAD_ASYNC_TO_LDS_B128 (W always;
//   A unless it must fuse the GIN add), drained with s_wait_asynccnt so the
//   next stage's traffic hides under the WMMA stream. The conv1 K tail is a
//   clamp+select on the A side only (WT pad columns are zero).
// Requires M % 64 == 0 (holds: 102400, 512) and Npad % 64 == 0.
// ---------------------------------------------------------------------------
template <bool HAS_A2, bool KTAIL, bool NTAIL>
__global__ __launch_bounds__(128) void wmma_gemm_kernel(
    const float* __restrict__ A1, const float* __restrict__ A2,
    const float* __restrict__ WT, const float* __restrict__ bias,
    float* __restrict__ Out, int M, int K, int N, int Kpad,
    int k_chunk, int use_atomic, int do_relu)
{
  // Row stride 20 floats: float4-aligned rows, conflict-free column reads.
  __shared__ float As[2][64][20];
  __shared__ float Bs[2][64][20];

  constexpr bool A_ASYNC = !HAS_A2;   // async copy cannot fuse the GIN add
  constexpr int  NPER    = (A_ASYNC ? 2 : 0) + 2;   // async ops per stage

  const int tid  = threadIdx.x;
  const int wave = tid >> 5;
  const int lane = tid & 31;
  const int lm   = lane & 15;
  const int lh   = lane >> 4;

  const int nBase = blockIdx.x * 64;
  const int mBase = blockIdx.y * 64;
  const int kBeg  = blockIdx.z * k_chunk;
  int kEnd = kBeg + k_chunk;
  if (kEnd > K) kEnd = K;
  const int nCh = (kEnd - kBeg + 15) >> 4;

  v8f acc0 = {}, acc1 = {}, acc2 = {}, acc3 = {};

  // ---- stage one 16-wide K chunk into LDS buffer `buf` -------------------
  auto stage = [&](int buf, int k0, bool tail) {
    // A tile (64 rows x 16 k)
    #pragma unroll
    for (int it = 0; it < 2; ++it) {
      const int i = tid + it * 128;          // 256 float4 slots
      const int r = i >> 2, c4 = i & 3;
      if (A_ASYNC) {
        async_copy_f4(A1 + (size_t)(mBase + r) * K + k0 + c4 * 4,
                      &As[buf][r][c4 * 4]);
      } else if (!KTAIL || !tail) {
        const size_t base = (size_t)(mBase + r) * K + k0 + c4 * 4;
        float4 v = *reinterpret_cast<const float4*>(A1 + base);
        const float4 u = *reinterpret_cast<const float4*>(A2 + base);
        v.x += u.x; v.y += u.y; v.z += u.z; v.w += u.w;
        *reinterpret_cast<float4*>(&As[buf][r][c4 * 4]) = v;
        __builtin_prefetch(A1 + base + 16, 0, 1);
        __builtin_prefetch(A2 + base + 16, 0, 1);
      } else {
        // clamp: load stays in-bounds, zero past kEnd (WT pad is zero too)
        const size_t rowb = (size_t)(mBase + r) * K;
        #pragma unroll
        for (int u = 0; u < 4; ++u) {
          const int gk = k0 + c4 * 4 + u;
          const int ck = gk < K ? gk : K - 1;
          float v = A1[rowb + ck] + A2[rowb + ck];
          As[buf][r][c4 * 4 + u] = (gk < kEnd) ? v : 0.0f;
        }
      }
    }
    // B tile (64 n-rows x 16 k) from padded WT: always async, no guards.
    #pragma unroll
    for (int it = 0; it < 2; ++it) {
      const int i = tid + it * 128;
      const int r = i >> 2, c4 = i & 3;
      async_copy_f4(WT + (size_t)(nBase + r) * Kpad + k0 + c4 * 4,
                    &Bs[buf][r][c4 * 4]);
    }
  };

  // ---- 16 x V_WMMA_F32_16X16X4_F32 per stage per wave --------------------
  //  A frag: lane l, vgpr v -> M = l%16, K = kk + 2*(l/16) + v
  //  B frag: lane l, vgpr v -> N = l%16, K = kk + 2*(l/16) + v
  auto compute = [&](int buf) {
    #pragma unroll
    for (int kk = 0; kk < 16; kk += 4) {
      const int kr = kk + 2 * lh;
      v2f a, b0, b1, b2, b3;
      a.x  = As[buf][wave * 16 + lm][kr + 0];
      a.y  = As[buf][wave * 16 + lm][kr + 1];
      b0.x = Bs[buf][ 0 + lm][kr + 0];  b0.y = Bs[buf][ 0 + lm][kr + 1];
      b1.x = Bs[buf][16 + lm][kr + 0];  b1.y = Bs[buf][16 + lm][kr + 1];
      b2.x = Bs[buf][32 + lm][kr + 0];  b2.y = Bs[buf][32 + lm][kr + 1];
      b3.x = Bs[buf][48 + lm][kr + 0];  b3.y = Bs[buf][48 + lm][kr + 1];
      acc0 = __builtin_amdgcn_wmma_f32_16x16x4_f32(false, a, false, b0, (short)0, acc0, false, false);
      acc1 = __builtin_amdgcn_wmma_f32_16x16x4_f32(false, a, false, b1, (short)0, acc1, false, false);
      acc2 = __builtin_amdgcn_wmma_f32_16x16x4_f32(false, a, false, b2, (short)0, acc2, false, false);
      acc3 = __builtin_amdgcn_wmma_f32_16x16x4_f32(false, a, false, b3, (short)0, acc3, false, false);
    }
  };

  // ---- software pipeline over K chunks -----------------------------------
  int cur = 0;
  if (nCh > 0) stage(0, kBeg, nCh == 1);
  for (int c = 0; c < nCh; ++c) {
    const bool more = (c + 1) < nCh;
    if (more) stage(cur ^ 1, kBeg + 16 * (c + 1), (c + 2) == nCh);
    if (more) wait_asynccnt<NPER>();   // current stage landed; next in flight
    else      wait_asynccnt<0>();      // drain
    __syncthreads();
    compute(cur);
    __syncthreads();
    cur ^= 1;
  }

  // ---- epilogue. C/D layout: vgpr r, lane l -> m = r + 8*(l/16), n = l%16.
  const int mTile = mBase + wave * 16 + lh * 8;
  #pragma unroll
  for (int j = 0; j < 4; ++j) {
    const v8f acc = (j == 0) ? acc0 : (j == 1) ? acc1 : (j == 2) ? acc2 : acc3;
    const int n = nBase + j * 16 + lm;
    if (NTAIL && n >= N) continue;
    if (use_atomic) {
      #pragma unroll
      for (int r = 0; r < 8; ++r)
        __hip_atomic_fetch_add(&Out[(size_t)(mTile + r) * N + n], acc[r],
                               __ATOMIC_RELAXED, __HIP_MEMORY_SCOPE_AGENT);
    } else {
      const float bv = bias ? bias[n] : 0.0f;
      #pragma unroll
      for (int r = 0; r < 8; ++r) {
        float v = acc[r] + bv;
        if (do_relu) v = fmaxf(v, 0.0f);
        Out[(size_t)(mTile + r) * N + n] = v;
      }
    }
  }
}

// ---------------------------------------------------------------------------
// Post split-K epilogue for fc1: y = relu(BN(y + fc1_b))
// ---------------------------------------------------------------------------
__global__ void bias_bn_relu_kernel(float* __restrict__ y,
                                    const float* __restrict__ bias,
                                    const float* __restrict__ mean,
                                    const float* __restrict__ var,
                                    const float* __restrict__ gamma,
                                    const float* __restrict__ beta,
                                    int total, int N) {
  int i = blockIdx.x * blockDim.x + threadIdx.x;
  if (i >= total) return;
  int n = i % N;
  float v = y[i] + bias[n];
  v = (v - mean[n]) * rsqrtf(var[n] + BN_EPS_C) * gamma[n] + beta[n];
  y[i] = fmaxf(v, 0.0f);
}

// ---------------------------------------------------------------------------
// fc2: logits[512,2] = y[512,200] @ w[200,2] + b   (tiny; one thread/output)
// ---------------------------------------------------------------------------
__global__ void fc2_kernel(const float* __restrict__ y,
                           const float* __restrict__ w,
                           const float* __restrict__ b,
                           float* __restrict__ out) {
  int i = blockIdx.x * blockDim.x + threadIdx.x;
  if (i >= BATCH_C * 2) return;
  int bi = i >> 1, c = i & 1;
  float acc = b[c];
  #pragma unroll 4
  for (int k = 0; k < NUM_ROI_C; ++k)
    acc += y[bi * NUM_ROI_C + k] * w[k * 2 + c];
  out[i] = acc;
}

// ---------------------------------------------------------------------------
extern "C" void kernel_launch(void* const* d_in, const int* in_sizes, int n_in,
                              void* d_out, int out_size, void* d_ws, size_t ws_size,
                              hipStream_t stream) {
  const float* x     = (const float*)d_in[0];
  const int*   ei    = (const int*)d_in[1];     // [2, N_EDGES]
  const float* w1a   = (const float*)d_in[2];
  const float* b1a   = (const float*)d_in[3];
  const float* w1b   = (const float*)d_in[4];
  const float* b1b   = (const float*)d_in[5];
  const float* w2a   = (const float*)d_in[6];
  const float* b2a   = (const float*)d_in[7];
  const float* w2b   = (const float*)d_in[8];
  const float* b2b   = (const float*)d_in[9];
  const float* fc1_w = (const float*)d_in[10];
  const float* fc1_b = (const float*)d_in[11];
  const float* bn_g  = (const float*)d_in[12];
  const float* bn_b  = (const float*)d_in[13];
  const float* bn_m  = (const float*)d_in[14];
  const float* bn_v  = (const float*)d_in[15];
  const float* fc2_w = (const float*)d_in[16];
  const float* fc2_b = (const float*)d_in[17];

  const int* src = ei;
  const int* dst = ei + N_EDGES_C;

  // Padded transposed-weight geometry
  const int K1PAD = 208;                       // conv1 K=200 -> 208
  const int KFC1  = NUM_ROI_C * HID_C;         // 25600 (already mult of 16)
  const int NFC1P = 256;                       // fc1 N=200 -> 256 rows

  // Workspace layout (floats), liveness-based reuse (~266 MB total):
  float* ws    = (float*)d_ws;
  float* agg1  = ws;                                        // N*200
  float* h1a   = agg1  + (size_t)N_NODES_C * NUM_ROI_C;     // N*128
  float* h1    = h1a   + (size_t)N_NODES_C * HID_C;         // N*128
  float* agg2  = h1    + (size_t)N_NODES_C * HID_C;         // N*128
  float* wt1a  = agg2  + (size_t)N_NODES_C * HID_C;         // 128 x 208
  float* wt1b  = wt1a  + (size_t)HID_C * K1PAD;             // 128 x 128
  float* wt2a  = wt1b  + (size_t)HID_C * HID_C;             // 128 x 128
  float* wt2b  = wt2a  + (size_t)HID_C * HID_C;             // 128 x 128
  float* wtfc1 = wt2b  + (size_t)HID_C * HID_C;             // 256 x 25600
  float* h2a   = h1a;    // h1a dead after gemm1b
  float* h2    = agg1;   // agg1 dead after gemm1a
  float* y     = agg2;   // agg2 dead after gemm2a; y = 512*200
  float* out   = (float*)d_out;

  const dim3 blk256(256), blk128(128), blkT(32, 8);

  // ---- one-time weight transposes (cheap; ~21 MB total) ------------------
  transpose_pad_kernel<<<dim3((K1PAD + 31) / 32, HID_C / 32), blkT, 0, stream>>>(
      w1a, wt1a, NUM_ROI_C, HID_C, K1PAD, HID_C);
  transpose_pad_kernel<<<dim3(HID_C / 32, HID_C / 32), blkT, 0, stream>>>(
      w1b, wt1b, HID_C, HID_C, HID_C, HID_C);
  transpose_pad_kernel<<<dim3(HID_C / 32, HID_C / 32), blkT, 0, stream>>>(
      w2a, wt2a, HID_C, HID_C, HID_C, HID_C);
  transpose_pad_kernel<<<dim3(HID_C / 32, HID_C / 32), blkT, 0, stream>>>(
      w2b, wt2b, HID_C, HID_C, HID_C, HID_C);
  transpose_pad_kernel<<<dim3(KFC1 / 32, NFC1P / 32), blkT, 0, stream>>>(
      fc1_w, wtfc1, KFC1, NUM_ROI_C, KFC1, NFC1P);

  // ---- GINConv 1 ----------------------------------------------------------
  {
    long long n = (long long)N_NODES_C * NUM_ROI_C;
    fill_zero_kernel<<<dim3((unsigned)((n + 255) / 256)), blk256, 0, stream>>>(agg1, n);
    long long nwork = (long long)N_EDGES_C * (NUM_ROI_C / 4);
    scatter_add_kernel<<<dim3((unsigned)((nwork + 255) / 256)), blk256, 0, stream>>>(
        x, src, dst, agg1, NUM_ROI_C, nwork);
  }
  // h1a = relu((x + agg1) @ w1a + b1a)   [102400 x 200] x [200 x 128]
  wmma_gemm_kernel<true, true, false>
      <<<dim3(HID_C / 64, N_NODES_C / 64, 1), blk128, 0, stream>>>(
      x, agg1, wt1a, b1a, h1a, N_NODES_C, NUM_ROI_C, HID_C, K1PAD,
      NUM_ROI_C, /*atomic=*/0, /*relu=*/1);
  // h1 = relu(h1a @ w1b + b1b)           [102400 x 128] x [128 x 128]
  wmma_gemm_kernel<false, false, false>
      <<<dim3(HID_C / 64, N_NODES_C / 64, 1), blk128, 0, stream>>>(
      h1a, nullptr, wt1b, b1b, h1, N_NODES_C, HID_C, HID_C, HID_C,
      HID_C, 0, 1);

  // ---- GINConv 2 ----------------------------------------------------------
  {
    long long n = (long long)N_NODES_C * HID_C;
    fill_zero_kernel<<<dim3((unsigned)((n + 255) / 256)), blk256, 0, stream>>>(agg2, n);
    long long nwork = (long long)N_EDGES_C * (HID_C / 4);
    scatter_add_kernel<<<dim3((unsigned)((nwork + 255) / 256)), blk256, 0, stream>>>(
        h1, src, dst, agg2, HID_C, nwork);
  }
  wmma_gemm_kernel<true, false, false>
      <<<dim3(HID_C / 64, N_NODES_C / 64, 1), blk128, 0, stream>>>(
      h1, agg2, wt2a, b2a, h2a, N_NODES_C, HID_C, HID_C, HID_C, HID_C, 0, 1);
  wmma_gemm_kernel<false, false, false>
      <<<dim3(HID_C / 64, N_NODES_C / 64, 1), blk128, 0, stream>>>(
      h2a, nullptr, wt2b, b2b, h2, N_NODES_C, HID_C, HID_C, HID_C, HID_C, 0, 1);

  // ---- fc1 + BN + ReLU ----------------------------------------------------
  // h2 memory is already [512, 25600] row-major (reshape is a no-op).
  {
    long long n = (long long)BATCH_C * NUM_ROI_C;
    fill_zero_kernel<<<dim3((unsigned)((n + 255) / 256)), blk256, 0, stream>>>(y, n);
    const int kSplit = 16;
    const int kChunk = KFC1 / kSplit;        // 1600, multiple of 16
    wmma_gemm_kernel<false, false, true>
        <<<dim3(NFC1P / 64, BATCH_C / 64, kSplit), blk128, 0, stream>>>(
        h2, nullptr, wtfc1, nullptr, y, BATCH_C, KFC1, NUM_ROI_C, KFC1,
        kChunk, /*atomic=*/1, /*relu=*/0);
    bias_bn_relu_kernel<<<dim3((unsigned)((n + 255) / 256)), blk256, 0, stream>>>(
        y, fc1_b, bn_m, bn_v, bn_g, bn_b, (int)n, NUM_ROI_C);
  }

  // ---- fc2 ----------------------------------------------------------------
  fc2_kernel<<<dim3((BATCH_C * 2 + 255) / 256), blk256, 0, stream>>>(
      y, fc2_w, fc2_b, out);
}